// SAGE_Encoder_41059887350178
// MI455X (gfx1250) — compile-verified
//
#include <hip/hip_runtime.h>
#include <hip/hip_bf16.h>

// ---------------------------------------------------------------------------
// SAGE encoder for MI455X (gfx1250, wave32).
//   - edge aggregation: wave-per-edge, f32 L2 atomics (all aggregates fit in
//     the 192MB L2, so atomic RMW stays on-chip)
//   - dense part: fp32 WMMA (v_wmma_f32_16x16x4_f32). Block = 16 waves,
//     computes 64 rows x 256 cols; each B fragment feeds 4 WMMAs.
//   - A tiles staged with CDNA5 async global->LDS b128 copies (ASYNCcnt),
//     mean division hoisted to one precise divide per row (LDS-cached).
// ---------------------------------------------------------------------------

#define N_USER  100000
#define N_MOVIE 20000
#define N_EDGES 1000000
#define DIM     256
#define ROWS    64            // rows per block
#define LDS_PAD 260           // floats per LDS row: 16B-aligned, bank-spread

typedef __attribute__((ext_vector_type(2))) float v2f;
typedef __attribute__((ext_vector_type(8))) float v8f;

// ------------------------- edge scatter (aggregation) ----------------------
// one wave per edge; lane l covers dims [8l, 8l+8)
__global__ __launch_bounds__(256)
void sage_scatter(const float* __restrict__ xsrc,
                  const int*   __restrict__ src,
                  const int*   __restrict__ dst,
                  const float* __restrict__ w,
                  float* __restrict__ agg,
                  float* __restrict__ cnt,
                  int n_edges) {
    const int gid  = blockIdx.x * blockDim.x + threadIdx.x;
    const int e    = gid >> 5;
    const int lane = threadIdx.x & 31;
    if (e >= n_edges) return;

    const int   s  = src[e];
    const int   d  = dst[e];
    const float wt = w[e];

    const float4* xp = (const float4*)(xsrc + (size_t)s * DIM + lane * 8);
    float4 v0 = xp[0];
    float4 v1 = xp[1];

    float* ap = agg + (size_t)d * DIM + lane * 8;
    atomicAdd(ap + 0, v0.x * wt);
    atomicAdd(ap + 1, v0.y * wt);
    atomicAdd(ap + 2, v0.z * wt);
    atomicAdd(ap + 3, v0.w * wt);
    atomicAdd(ap + 4, v1.x * wt);
    atomicAdd(ap + 5, v1.y * wt);
    atomicAdd(ap + 6, v1.z * wt);
    atomicAdd(ap + 7, v1.w * wt);
    if (lane == 0) atomicAdd(cnt + d, 1.0f);
}

// ---------------------- CDNA5 async global->LDS helpers --------------------
__device__ __forceinline__ unsigned lds_off(const void* p) {
    // generic pointers to LDS carry the LDS byte offset in the low 32 bits
    return (unsigned)(unsigned long long)p;
}

__device__ __forceinline__ void async_copy16(unsigned lds, const float* g) {
    asm volatile("global_load_async_to_lds_b128 %0, %1, off"
                 :: "v"(lds), "v"(g) : "memory");
}

__device__ __forceinline__ void wait_async_all() {
    asm volatile("s_wait_asynccnt 0x0" ::: "memory");
}

// ---------------- fused dual-GEMM + mean + bias (+relu+residual) -----------
// out = (agg@Wl)*(1/max(cnt,1)) + xdst@Wr + b         (layer 2)
// out = xdst + relu(above)                            (layer 1, RELU_RES)
template <bool RELU_RES>
__global__ __launch_bounds__(512)
void sage_gemm(const float* __restrict__ agg,
               const float* __restrict__ cnt,
               const float* __restrict__ xdst,
               const float* __restrict__ Wl,
               const float* __restrict__ Wr,
               const float* __restrict__ bias,
               float* __restrict__ out,
               int n) {
    __shared__ float sm[ROWS][LDS_PAD];   // raw agg tile (mean deferred)
    __shared__ float sx[ROWS][LDS_PAD];   // xdst tile
    __shared__ float ssc[ROWS];           // per-row 1/max(cnt,1)

    const int r0   = blockIdx.x * ROWS;
    const int t    = threadIdx.x;
    const int wave = t >> 5;
    const int lane = t & 31;

    // ---- stage A tiles with async b128 copies (row-clamped, in-bounds) ----
    for (int idx = t; idx < ROWS * (DIM / 4); idx += 512) {
        const int row = idx >> 6;                 // DIM/4 == 64 chunks/row
        const int ch  = idx & 63;                 // 16B chunk
        int gr = r0 + row;
        gr = (gr < n) ? gr : (n - 1);
        async_copy16(lds_off(&sm[row][ch * 4]), agg  + (size_t)gr * DIM + ch * 4);
        async_copy16(lds_off(&sx[row][ch * 4]), xdst + (size_t)gr * DIM + ch * 4);
    }
    // one precise divide per row (instead of 32 per lane in the epilogue)
    if (t < ROWS) {
        int gr = r0 + t;
        gr = (gr < n) ? gr : (n - 1);
        ssc[t] = 1.0f / fmaxf(cnt[gr], 1.0f);
    }
    wait_async_all();
    __syncthreads();

    // ---- WMMA K-loop ----
    // A frag (16x4 f32): lane L -> m = L%16, holds K = 2*(L/16)+{0,1}
    // B frag (4x16 f32): lane L -> n = L%16, holds K = 2*(L/16)+{0,1}
    const int c0 = wave * 16;
    const int m  = lane & 15;
    const int kp = (lane >> 4) * 2;

    v8f accL[4] = {};
    v8f accR[4] = {};
    for (int k = 0; k < DIM; k += 4) {
        // keep the L2-resident weight stream ahead (wrap keeps it in-bounds)
        const int kpre = (k + 16) & (DIM - 1);
        __builtin_prefetch(&Wl[(size_t)(kpre + kp) * DIM + c0 + m], 0, 1);
        __builtin_prefetch(&Wr[(size_t)(kpre + kp) * DIM + c0 + m], 0, 1);

        v2f b1, b2;
        b1.x = Wl[(size_t)(k + kp) * DIM + c0 + m];
        b1.y = Wl[(size_t)(k + kp + 1) * DIM + c0 + m];
        b2.x = Wr[(size_t)(k + kp) * DIM + c0 + m];
        b2.y = Wr[(size_t)(k + kp + 1) * DIM + c0 + m];

#pragma unroll
        for (int rt = 0; rt < 4; ++rt) {
            const v2f a1 = *(const v2f*)&sm[rt * 16 + m][k + kp];  // 8B aligned
            const v2f a2 = *(const v2f*)&sx[rt * 16 + m][k + kp];
            accL[rt] = __builtin_amdgcn_wmma_f32_16x16x4_f32(
                false, a1, false, b1, (short)0, accL[rt], false, false);
            accR[rt] = __builtin_amdgcn_wmma_f32_16x16x4_f32(
                false, a2, false, b2, (short)0, accR[rt], false, false);
        }
    }

    // ---- epilogue: mean scale + bias (+ relu + residual) ----
    // D frag: lane L -> n = L%16, VGPR v -> row = v + 8*(L/16)
    const int  half = lane >> 4;
    const int  ncol = c0 + m;
    const float bv  = bias[ncol];

    if (r0 + ROWS <= n) {
        // common case: whole block in-bounds, no per-element masking
#pragma unroll
        for (int rt = 0; rt < 4; ++rt) {
#pragma unroll
            for (int v = 0; v < 8; ++v) {
                const int row = rt * 16 + v + 8 * half;
                float h = accL[rt][v] * ssc[row] + accR[rt][v] + bv;
                if (RELU_RES) h = sx[row][ncol] + fmaxf(h, 0.0f);
                out[(size_t)(r0 + row) * DIM + ncol] = h;
            }
        }
    } else {
#pragma unroll
        for (int rt = 0; rt < 4; ++rt) {
#pragma unroll
            for (int v = 0; v < 8; ++v) {
                const int row = rt * 16 + v + 8 * half;
                const int gr  = r0 + row;
                if (gr < n) {
                    float h = accL[rt][v] * ssc[row] + accR[rt][v] + bv;
                    if (RELU_RES) h = sx[row][ncol] + fmaxf(h, 0.0f);
                    out[(size_t)gr * DIM + ncol] = h;
                }
            }
        }
    }
}

// ---------------------------------------------------------------------------
extern "C" void kernel_launch(void* const* d_in, const int* in_sizes, int n_in,
                              void* d_out, int out_size, void* d_ws, size_t ws_size,
                              hipStream_t stream) {
    const float* x_user  = (const float*)d_in[0];
    const float* x_movie = (const float*)d_in[1];
    const int*   src_um  = (const int*)d_in[2];
    const int*   dst_um  = (const int*)d_in[3];
    const float* w_um    = (const float*)d_in[4];
    const int*   src_mu  = (const int*)d_in[5];
    const int*   dst_mu  = (const int*)d_in[6];
    const float* w_mu    = (const float*)d_in[7];
    const float* c1_um_Wl = (const float*)d_in[8];
    const float* c1_um_Wr = (const float*)d_in[9];
    const float* c1_um_b  = (const float*)d_in[10];
    const float* c1_mu_Wl = (const float*)d_in[11];
    const float* c1_mu_Wr = (const float*)d_in[12];
    const float* c1_mu_b  = (const float*)d_in[13];
    const float* c2_um_Wl = (const float*)d_in[14];
    const float* c2_um_Wr = (const float*)d_in[15];
    const float* c2_um_b  = (const float*)d_in[16];
    const float* c2_mu_Wl = (const float*)d_in[17];
    const float* c2_mu_Wr = (const float*)d_in[18];
    const float* c2_mu_b  = (const float*)d_in[19];

    // workspace layout (floats):
    // [agg_movie | cnt_movie | agg_user | cnt_user | r_user | r_movie]
    float* ws        = (float*)d_ws;
    float* agg_movie = ws;
    float* cnt_movie = agg_movie + (size_t)N_MOVIE * DIM;
    float* agg_user  = cnt_movie + N_MOVIE;
    float* cnt_user  = agg_user + (size_t)N_USER * DIM;
    float* r_user    = cnt_user + N_USER;
    float* r_movie   = r_user + (size_t)N_USER * DIM;

    const size_t zero_bytes =
        ((size_t)N_MOVIE * DIM + N_MOVIE + (size_t)N_USER * DIM + N_USER) * sizeof(float);

    float* out_user  = (float*)d_out;
    float* out_movie = out_user + (size_t)N_USER * DIM;

    const dim3 sc_blk(256);
    const dim3 sc_grd((N_EDGES * 32 + 255) / 256);
    const dim3 gm_blk(512);
    const dim3 gm_grd_user((N_USER + ROWS - 1) / ROWS);
    const dim3 gm_grd_movie((N_MOVIE + ROWS - 1) / ROWS);

    // ---------------- layer 1 ----------------
    hipMemsetAsync(d_ws, 0, zero_bytes, stream);
    sage_scatter<<<sc_grd, sc_blk, 0, stream>>>(x_user, src_um, dst_um, w_um,
                                                agg_movie, cnt_movie, N_EDGES);
    sage_scatter<<<sc_grd, sc_blk, 0, stream>>>(x_movie, src_mu, dst_mu, w_mu,
                                                agg_user, cnt_user, N_EDGES);
    sage_gemm<true><<<gm_grd_movie, gm_blk, 0, stream>>>(
        agg_movie, cnt_movie, x_movie, c1_um_Wl, c1_um_Wr, c1_um_b, r_movie, N_MOVIE);
    sage_gemm<true><<<gm_grd_user, gm_blk, 0, stream>>>(
        agg_user, cnt_user, x_user, c1_mu_Wl, c1_mu_Wr, c1_mu_b, r_user, N_USER);

    // ---------------- layer 2 ----------------
    hipMemsetAsync(d_ws, 0, zero_bytes, stream);
    sage_scatter<<<sc_grd, sc_blk, 0, stream>>>(r_user, src_um, dst_um, w_um,
                                                agg_movie, cnt_movie, N_EDGES);
    sage_scatter<<<sc_grd, sc_blk, 0, stream>>>(r_movie, src_mu, dst_mu, w_mu,
                                                agg_user, cnt_user, N_EDGES);
    sage_gemm<false><<<gm_grd_user, gm_blk, 0, stream>>>(
        agg_user, cnt_user, r_user, c2_mu_Wl, c2_mu_Wr, c2_mu_b, out_user, N_USER);
    sage_gemm<false><<<gm_grd_movie, gm_blk, 0, stream>>>(
        agg_movie, cnt_movie, r_movie, c2_um_Wl, c2_um_Wr, c2_um_b, out_movie, N_MOVIE);
}